// AcRKNLayer_63952063038113
// MI455X (gfx1250) — compile-verified
//
#include <hip/hip_runtime.h>
#include <hip/hip_bf16.h>
#include <math.h>

// Problem constants (from reference)
#define B_   512
#define T_   256
#define LOD_ 64
#define LSD_ 128
#define ACT_ 32
#define K_   15
#define H_   120

typedef __attribute__((ext_vector_type(2))) float v2f;
typedef __attribute__((ext_vector_type(8))) float v8f;

// fp32 WMMA: D(16x16) = A(16x4) * B(4x16) + C, wave32.
// A layout: lane L (g=L>>4,h=L&15): row=h, K = kc + 2g + v  (v = vector elem 0/1)
// B layout: col=h, K = kc + 2g + v
// C/D layout: row = v + 8g (v=0..7), col = h
__device__ __forceinline__ v8f wmma_f32(v2f a, v2f b, v8f c) {
    return __builtin_amdgcn_wmma_f32_16x16x4_f32(false, a, false, b, (short)0, c,
                                                 false, false);
}

// Async global -> LDS copy of 16 bytes per lane (CDNA5 GLOBAL_LOAD_ASYNC_TO_LDS).
// Tracked by ASYNCcnt; completes in-order with other async loads of this wave.
__device__ __forceinline__ void async_copy_b128(unsigned lds_off, const float* gsrc) {
    asm volatile("global_load_async_to_lds_b128 %0, %1, off"
                 :: "v"(lds_off), "v"(gsrc)
                 : "memory");
}

template <int N>
__device__ __forceinline__ void wait_asynccnt() {
    asm volatile("s_wait_asynccnt %0" :: "i"(N) : "memory");
}

// Generic pointers to LDS are aperture-based: low 32 bits == LDS byte offset.
__device__ __forceinline__ unsigned lds_offset(const void* p) {
    return (unsigned)(unsigned long long)p;
}

// ---------------------------------------------------------------------------
// Kernel 1: ctrl[b,t,:] = relu(act @ w1 + b1) @ w2 + b2, written into the
// prior_means output region (kernel 2 reads it at step t, then overwrites it).
// Block = 128 threads (4 waves), each wave owns 16 rows of flattened (B*T).
// Throughput phase: 8192 independent waves, global-load latency fully hidden.
// ---------------------------------------------------------------------------
__global__ __launch_bounds__(128) void ackrn_ctrl_kernel(
    const float* __restrict__ actions,   // (B*T, 32)
    const float* __restrict__ w1,        // (32, 120)
    const float* __restrict__ b1,        // (120,)
    const float* __restrict__ w2,        // (120, 128)
    const float* __restrict__ b2,        // (128,)
    float* __restrict__ ctrl_out)        // (B*T, 128)  == prior_means region
{
    __shared__ float hid[4][16][132];    // padded stride vs 64 banks

    const int wave = threadIdx.x >> 5;
    const int lane = threadIdx.x & 31;
    const int g = lane >> 4;
    const int h = lane & 15;
    const long row0 = (long)blockIdx.x * 64 + (long)wave * 16;

    // ---- hidden = relu(act @ w1 + b1), padded to 16x128 in LDS ----
    for (int nt = 0; nt < 8; ++nt) {
        v8f acc = {};
        #pragma unroll
        for (int kc = 0; kc < ACT_; kc += 4) {
            const int k0 = kc + 2 * g;
            v2f a, b;
            const float* ar = actions + (row0 + h) * ACT_ + k0;
            a[0] = ar[0];
            a[1] = ar[1];
            const int n = nt * 16 + h;
            const bool nok = (n < H_);
            b[0] = nok ? w1[k0 * H_ + n] : 0.0f;
            b[1] = nok ? w1[(k0 + 1) * H_ + n] : 0.0f;
            acc = wmma_f32(a, b, acc);
        }
        const int n = nt * 16 + h;
        const float bias = (n < H_) ? b1[n] : 0.0f;
        #pragma unroll
        for (int v = 0; v < 8; ++v) {
            float x = acc[v] + bias;
            hid[wave][v + 8 * g][n] = x > 0.0f ? x : 0.0f;
        }
    }
    __syncthreads();

    // ---- ctrl = hidden @ w2 + b2 (K = 120 exactly, no pad reads) ----
    for (int nt = 0; nt < 8; ++nt) {
        v8f acc = {};
        for (int kc = 0; kc < H_; kc += 4) {
            const int k0 = kc + 2 * g;
            v2f a, b;
            a[0] = hid[wave][h][k0];
            a[1] = hid[wave][h][k0 + 1];
            const int n = nt * 16 + h;
            b[0] = w2[k0 * LSD_ + n];
            b[1] = w2[(k0 + 1) * LSD_ + n];
            acc = wmma_f32(a, b, acc);
        }
        const int n = nt * 16 + h;
        const float bias = b2[n];
        #pragma unroll
        for (int v = 0; v < 8; ++v) {
            ctrl_out[(row0 + v + 8 * g) * LSD_ + n] = acc[v] + bias;
        }
    }
}

// ---------------------------------------------------------------------------
// Kernel 2: the sequential Kalman scan. One block = 16 batch rows, 8 waves.
// All recurrent state + all small constant matrices live in LDS; observation
// tiles are double-buffered via async global->LDS DMA so HBM latency overlaps
// a full step of WMMA/softmax work on the serial critical path.
// ---------------------------------------------------------------------------
__global__ __launch_bounds__(256) void ackrn_scan_kernel(
    const float* __restrict__ latent_obs,   // (B,T,64)
    const float* __restrict__ obs_vars,     // (B,T,64)
    const unsigned char* __restrict__ obs_valid, // (B,T) bool
    const float* __restrict__ initial_mean, // (B,128)
    const float* __restrict__ initial_cov,  // (B,3,64)
    const float* __restrict__ tm11,         // (15,64)
    const float* __restrict__ tm12,
    const float* __restrict__ tm21,
    const float* __restrict__ tm22,
    const float* __restrict__ w_coeff,      // (128,15)
    const float* __restrict__ b_coeff,      // (15,)
    const float* __restrict__ log_trans_noise, // (128,)
    float* __restrict__ post_means,         // (B,T,128)
    float* __restrict__ post_covs,          // (B,T,3,64)
    float* __restrict__ prior_means,        // (B,T,128)  holds ctrl on entry
    float* __restrict__ post_covs2)         // duplicate of post_covs
{
    __shared__ float mu[16][132];
    __shared__ float pmu[16][132];
    __shared__ float cu[16][68], cl[16][68], cs[16][68];
    __shared__ float pcu[16][68], pcl[16][68], pcs[16][68];
    __shared__ float coeff[16][20];
    __shared__ float tmat[4][16][68];
    __shared__ float nu[64], nl[64];
    // constant operands, staged once (zero-padded: no guards in WMMA loops)
    __shared__ float wcs[128][20];          // w_coeff padded to 128x16
    __shared__ float tms[4][16][68];        // tm* padded to 16x64
    // double-buffered observation tiles (16 rows x 64), filled by async DMA
    __shared__ float obsb[2][16 * 64];
    __shared__ float varb[2][16 * 64];

    const int tid  = threadIdx.x;
    const int wave = tid >> 5;
    const int lane = tid & 31;
    const int g = lane >> 4;
    const int h = lane & 15;
    const int b0 = blockIdx.x * 16;

    // ---- initialize recurrent state ----
    for (int i = tid; i < 16 * LSD_; i += 256) {
        int r = i >> 7, c = i & 127;
        mu[r][c] = initial_mean[(long)(b0 + r) * LSD_ + c];
    }
    for (int i = tid; i < 16 * LOD_; i += 256) {
        int r = i >> 6, c = i & 63;
        const long base = (long)(b0 + r) * 3 * LOD_;
        cu[r][c] = initial_cov[base + c];
        cl[r][c] = initial_cov[base + LOD_ + c];
        cs[r][c] = initial_cov[base + 2 * LOD_ + c];
    }
    if (tid < LSD_) {
        float x = log_trans_noise[tid];
        float n = (x > 0.0f ? x : (__expf(x) - 1.0f)) + 1.0f; // elu(x)+1
        if (tid < LOD_) nu[tid] = n; else nl[tid - LOD_] = n;
    }
    // ---- stage constant matrices into LDS with zero padding ----
    for (int i = tid; i < 128 * 16; i += 256) {
        int k = i >> 4, n = i & 15;
        wcs[k][n] = (n < K_) ? w_coeff[k * K_ + n] : 0.0f;
    }
    for (int i = tid; i < 4 * 16 * 64; i += 256) {
        int m = i >> 10, k = (i >> 6) & 15, n = i & 63;
        const float* tm = (m == 0) ? tm11 : (m == 1) ? tm12
                         : (m == 2) ? tm21 : tm22;
        tms[m][k][n] = (k < K_) ? tm[k * LOD_ + n] : 0.0f;
    }

    // ---- prologue: async-DMA the t=0 observation tile into buffer 0 ----
    {
        const int r = tid >> 4;             // 16 rows
        const int j = (tid & 15) * 4;       // 16B chunk within row
        const long gbase = ((long)(b0 + r) * T_ + 0) * LOD_ + j;
        async_copy_b128(lds_offset(&obsb[0][tid * 4]), latent_obs + gbase);
        async_copy_b128(lds_offset(&varb[0][tid * 4]), obs_vars + gbase);
    }
    wait_asynccnt<0>();
    __syncthreads();

    for (int t = 0; t < T_; ++t) {
        const int buf = t & 1;

        // ---- kick off async DMA for step t+1 into the other buffer ----
        if (t + 1 < T_) {
            const int r = tid >> 4;
            const int j = (tid & 15) * 4;
            const long gbase = ((long)(b0 + r) * T_ + (t + 1)) * LOD_ + j;
            async_copy_b128(lds_offset(&obsb[buf ^ 1][tid * 4]),
                            latent_obs + gbase);
            async_copy_b128(lds_offset(&varb[buf ^ 1][tid * 4]),
                            obs_vars + gbase);
            // allow the 2 just-issued ops to remain in flight; the previous
            // step's 2 ops (in-order completion) are then guaranteed done
            wait_asynccnt<2>();
        } else {
            wait_asynccnt<0>();
        }
        __syncthreads();

        // ---- Phase A: Kalman observation update (16 rows x 64 cols) ----
        for (int i = tid; i < 16 * LOD_; i += 256) {
            const int r = i >> 6, j = i & 63;
            const long b = b0 + r;
            const float obs = obsb[buf][i];
            const float var = varb[buf][i];
            const bool valid = obs_valid[b * T_ + t] != 0;
            const float muu = mu[r][j], mul = mu[r][LOD_ + j];
            const float cuv = cu[r][j], clv = cl[r][j], csv = cs[r][j];
            const float denom = cuv + var;
            const float qu = cuv / denom;
            const float ql = csv / denom;
            const float res = obs - muu;
            const float pu = valid ? muu + qu * res : muu;
            const float pl = valid ? mul + ql * res : mul;
            const float u2 = valid ? (1.0f - qu) * cuv : cuv;
            const float l2 = valid ? clv - ql * csv : clv;
            const float s2 = valid ? (1.0f - qu) * csv : csv;
            pmu[r][j] = pu; pmu[r][LOD_ + j] = pl;
            pcu[r][j] = u2; pcl[r][j] = l2; pcs[r][j] = s2;
            const long ob = b * T_ + t;
            post_means[ob * LSD_ + j] = pu;
            post_means[ob * LSD_ + LOD_ + j] = pl;
            const long cvb = ob * 3 * LOD_;
            post_covs[cvb + j] = u2;
            post_covs[cvb + LOD_ + j] = l2;
            post_covs[cvb + 2 * LOD_ + j] = s2;
            post_covs2[cvb + j] = u2;
            post_covs2[cvb + LOD_ + j] = l2;
            post_covs2[cvb + 2 * LOD_ + j] = s2;
        }
        __syncthreads();

        // ---- Phase B: coeff logits = post_mu @ w_coeff + b_coeff (wave 0) ----
        if (wave == 0) {
            v8f acc = {};
            for (int kc = 0; kc < LSD_; kc += 4) {
                const int k0 = kc + 2 * g;
                v2f a, b;
                a[0] = pmu[h][k0];
                a[1] = pmu[h][k0 + 1];
                b[0] = wcs[k0][h];
                b[1] = wcs[k0 + 1][h];
                acc = wmma_f32(a, b, acc);
            }
            const float bc = (h < K_) ? b_coeff[h] : 0.0f;
            #pragma unroll
            for (int v = 0; v < 8; ++v)
                coeff[v + 8 * g][h] = acc[v] + bc;
        }
        __syncthreads();

        // ---- Phase C: softmax over the 15 valid logits, per row ----
        if (tid < 16) {
            float m = -1e30f;
            #pragma unroll
            for (int n = 0; n < K_; ++n) m = fmaxf(m, coeff[tid][n]);
            float s = 0.0f;
            float e[K_];
            #pragma unroll
            for (int n = 0; n < K_; ++n) { e[n] = __expf(coeff[tid][n] - m); s += e[n]; }
            const float inv = 1.0f / s;
            #pragma unroll
            for (int n = 0; n < K_; ++n) coeff[tid][n] = e[n] * inv;
            coeff[tid][15] = 0.0f;   // zero pad column feeding the K=16 WMMA loop
        }
        __syncthreads();

        // ---- Phase D: t{11,12,21,22} = coeff @ tm{..}; 16 tile jobs / 8 waves ----
        for (int jb = wave * 2; jb < wave * 2 + 2; ++jb) {
            const int m = jb >> 2;          // which transition matrix
            const int ntile = jb & 3;       // which 16-col tile of 64
            v8f acc = {};
            #pragma unroll
            for (int kc = 0; kc < 16; kc += 4) {
                const int k0 = kc + 2 * g;
                v2f a, b;
                a[0] = coeff[h][k0];
                a[1] = coeff[h][k0 + 1];
                const int n = ntile * 16 + h;
                b[0] = tms[m][k0][n];
                b[1] = tms[m][k0 + 1][n];
                acc = wmma_f32(a, b, acc);
            }
            #pragma unroll
            for (int v = 0; v < 8; ++v)
                tmat[m][v + 8 * g][ntile * 16 + h] = acc[v];
        }
        __syncthreads();

        // ---- Phase E: prediction step; ctrl read from prior region then
        //      overwritten with the new prior mean (same thread, same addrs) ----
        for (int i = tid; i < 16 * LOD_; i += 256) {
            const int r = i >> 6, j = i & 63;
            const long b = b0 + r;
            const long ob = b * T_ + t;
            const float T11 = tmat[0][r][j], T12 = tmat[1][r][j];
            const float T21 = tmat[2][r][j], T22 = tmat[3][r][j];
            const float pu = pmu[r][j], pl = pmu[r][LOD_ + j];
            const float u2 = pcu[r][j], l2 = pcl[r][j], s2 = pcs[r][j];
            const float ctrl_u = prior_means[ob * LSD_ + j];
            const float ctrl_l = prior_means[ob * LSD_ + LOD_ + j];
            const float nmu_u = T11 * pu + T12 * pl + ctrl_u;
            const float nmu_l = T21 * pu + T22 * pl + ctrl_l;
            const float ncu = T11 * T11 * u2 + 2.0f * T11 * T12 * s2
                            + T12 * T12 * l2 + nu[j];
            const float ncl = T21 * T21 * u2 + 2.0f * T21 * T22 * s2
                            + T22 * T22 * l2 + nl[j];
            const float ncs = T11 * T21 * u2 + (T11 * T22 + T12 * T21) * s2
                            + T12 * T22 * l2;
            mu[r][j] = nmu_u; mu[r][LOD_ + j] = nmu_l;
            cu[r][j] = ncu;   cl[r][j] = ncl;   cs[r][j] = ncs;
            prior_means[ob * LSD_ + j] = nmu_u;
            prior_means[ob * LSD_ + LOD_ + j] = nmu_l;
        }
        __syncthreads();
    }
}

// ---------------------------------------------------------------------------
extern "C" void kernel_launch(void* const* d_in, const int* in_sizes, int n_in,
                              void* d_out, int out_size, void* d_ws, size_t ws_size,
                              hipStream_t stream) {
    (void)in_sizes; (void)n_in; (void)out_size; (void)d_ws; (void)ws_size;

    const float* latent_obs      = (const float*)d_in[0];
    const float* obs_vars        = (const float*)d_in[1];
    const float* actions         = (const float*)d_in[2];
    const float* initial_mean    = (const float*)d_in[3];
    const float* initial_cov     = (const float*)d_in[4];
    const unsigned char* obs_valid = (const unsigned char*)d_in[5];
    const float* tm11            = (const float*)d_in[6];
    const float* tm12            = (const float*)d_in[7];
    const float* tm21            = (const float*)d_in[8];
    const float* tm22            = (const float*)d_in[9];
    const float* w_coeff         = (const float*)d_in[10];
    const float* b_coeff         = (const float*)d_in[11];
    const float* w1              = (const float*)d_in[12];
    const float* b1              = (const float*)d_in[13];
    const float* w2              = (const float*)d_in[14];
    const float* b2              = (const float*)d_in[15];
    const float* log_trans_noise = (const float*)d_in[16];

    float* out = (float*)d_out;
    const long n_pm = (long)B_ * T_ * LSD_;       // 16,777,216
    const long n_cv = (long)B_ * T_ * 3 * LOD_;   // 25,165,824
    float* post_means  = out;
    float* post_covs   = out + n_pm;
    float* prior_means = out + n_pm + n_cv;
    float* post_covs2  = out + 2 * n_pm + n_cv;

    // Phase 1: ctrl net for all (B,T) rows -> staged in prior_means region.
    ackrn_ctrl_kernel<<<(B_ * T_) / 64, 128, 0, stream>>>(
        actions, w1, b1, w2, b2, prior_means);

    // Phase 2: sequential scan; 32 blocks of 16 batch rows.
    ackrn_scan_kernel<<<B_ / 16, 256, 0, stream>>>(
        latent_obs, obs_vars, obs_valid, initial_mean, initial_cov,
        tm11, tm12, tm21, tm22, w_coeff, b_coeff, log_trans_noise,
        post_means, post_covs, prior_means, post_covs2);
}